// BlockDiagonal_30846455120439
// MI455X (gfx1250) — compile-verified
//
#include <hip/hip_runtime.h>

typedef float v2f __attribute__((ext_vector_type(2)));
typedef float v8f __attribute__((ext_vector_type(8)));

#define NTOT   8192     // full matrix dimension
#define BATCH  64
#define BLK    256      // diagonal block size
#define KHALF  128      // K range per wave (K split across 2 wave groups)
#define LDS_STRIDE 260  // 256 + 4 floats pad -> conflict-free column access

__launch_bounds__(512, 2)
__global__ void bdiag_wmma_f32(const float* __restrict__ x,
                               const float* __restrict__ blocks,
                               float* __restrict__ out) {
    __shared__ float aTile[BATCH * LDS_STRIDE];   // 66,560 B

    const int ib    = blockIdx.y;      // diagonal block (0..31)
    const int nhalf = blockIdx.x;      // 128-wide half of N (0/1)
    const int t     = threadIdx.x;
    const int wave  = t >> 5;          // 0..15
    const int lane  = t & 31;
    const int strip = wave & 7;        // 16-wide N strip within half
    const int khalf = wave >> 3;       // which K half this wave reduces

    const int colbase = ib * BLK;

    // ---- Stage x tile [64 x 256] into LDS; 4096 float4, 512 thr x 8 ----
    {
        const float4* xg = reinterpret_cast<const float4*>(x);
#pragma unroll
        for (int j = 0; j < 8; ++j) {
            const int f   = t + j * 512;
            const int row = f >> 6;
            const int c4  = f & 63;
            float4 v = xg[((size_t)row * NTOT + colbase) / 4 + c4];
            float* dst = &aTile[row * LDS_STRIDE + c4 * 4];
            dst[0] = v.x; dst[1] = v.y; dst[2] = v.z; dst[3] = v.w;
        }
    }
    __syncthreads();

    const int n0   = nhalf * 128 + strip * 16;
    const int nn   = lane & 15;
    const int kb   = (lane < 16) ? 0 : 2;   // lane-half K sub-offset
    const int colg = colbase + n0 + nn;     // global output column
    const int k0   = khalf * KHALF;         // this wave's K origin

    v8f acc[4] = {};                        // 4 M-tiles of 16x16 f32

    // B element (row colbase+k0+kb+k, col colg); row stride = NTOT floats
    const float* Bb = blocks + (size_t)(colbase + k0 + kb) * NTOT + colg;

    // ---- Depth-4 register pipeline for B fragments ----
    v2f breg[4];
#pragma unroll
    for (int p = 0; p < 4; ++p) {
        breg[p].x = Bb[(size_t)(4 * p) * NTOT];
        breg[p].y = Bb[(size_t)(4 * p + 1) * NTOT];
    }

#pragma unroll
    for (int k = 0; k < KHALF; k += 4) {     // fully unrolled: static guards
        const v2f b = breg[(k >> 2) & 3];
        if (k + 16 < KHALF) {                // refill slot 16 K-steps ahead
            breg[(k >> 2) & 3].x = Bb[(size_t)(k + 16) * NTOT];
            breg[(k >> 2) & 3].y = Bb[(size_t)(k + 17) * NTOT];
        }
        if (k + 24 < KHALF) {                // stream hint further ahead
            __builtin_prefetch(&Bb[(size_t)(k + 24) * NTOT], 0, 1);
        }

        const float* ap = &aTile[(lane & 15) * LDS_STRIDE + k0 + k + kb];
#pragma unroll
        for (int m = 0; m < 4; ++m) {
            v2f a;
            a.x = ap[m * 16 * LDS_STRIDE];
            a.y = ap[m * 16 * LDS_STRIDE + 1];
            // (neg_a, A, neg_b, B, c_mod, C, reuse_a, reuse_b)
            acc[m] = __builtin_amdgcn_wmma_f32_16x16x4_f32(
                false, a, false, b, (short)0, acc[m], false, false);
        }
    }

    // ---- Deterministic K reduction: high-K waves -> LDS, low-K adds ----
    __syncthreads();                         // everyone done reading aTile
    float* red = aTile;                      // reuse (32 KB needed)
    if (khalf == 1) {
#pragma unroll
        for (int m = 0; m < 4; ++m)
#pragma unroll
            for (int r = 0; r < 8; ++r)
                red[((strip * 4 + m) * 8 + r) * 32 + lane] = acc[m][r];
    }
    __syncthreads();
    if (khalf == 0) {
#pragma unroll
        for (int m = 0; m < 4; ++m) {
#pragma unroll
            for (int r = 0; r < 8; ++r) {
                const float v =
                    acc[m][r] + red[((strip * 4 + m) * 8 + r) * 32 + lane];
                const int row = m * 16 + ((lane < 16) ? r : (8 + r));
                out[(size_t)row * NTOT + colg] = v;
            }
        }
    }
}

extern "C" void kernel_launch(void* const* d_in, const int* in_sizes, int n_in,
                              void* d_out, int out_size, void* d_ws, size_t ws_size,
                              hipStream_t stream) {
    (void)in_sizes; (void)n_in; (void)d_ws; (void)ws_size; (void)out_size;
    const float* x      = (const float*)d_in[0];   // [64, 8192] f32
    const float* blocks = (const float*)d_in[1];   // [8192, 8192] f32
    // d_in[2] (mask) intentionally unused: block-diagonal structure is static.
    float* out = (float*)d_out;                    // [64, 8192] f32

    dim3 grid(2, 32, 1);    // 2 N-halves x 32 diagonal blocks
    dim3 block(512, 1, 1);  // 16 wave32s: 8 N strips x 2 K halves
    bdiag_wmma_f32<<<grid, block, 0, stream>>>(x, blocks, out);
}